// Net_50749333569766
// MI455X (gfx1250) — compile-verified
//
#include <hip/hip_runtime.h>
#include <hip/hip_bf16.h>
#include <math.h>

typedef __attribute__((ext_vector_type(2))) float v2f;
typedef __attribute__((ext_vector_type(8))) float v8f;

#define BN_EPS 1e-5f

// ---------------------------------------------------------------------------
// Implicit-GEMM 3x3 pad-1 conv using fp32 WMMA (V_WMMA_F32_16X16X4_F32).
// GEMM view: M = Cout, N = B*H*W, K = Cin*9. One wave owns a 16x16 D tile.
// A fragment (16x4 f32, 2 VGPR): lane L: M = L&15, K = 2*(L>>4)+{0,1}.
// C/D (8 VGPR): vgpr r, lane L -> M = r + 8*(L>>4), N = L&15.
// Fully branch-free K loop; all gathers use 32-bit element offsets from the
// wave-uniform base pointers (every tensor < 2^31 elements), so address math
// is a single 32-bit mad chain per tap and loads use SGPR-base addressing.
// Coordinates are clamped (address always valid) and only the loaded value
// is selected against zero. (ci,kk) is tracked incrementally; kk/3 uses the
// exact small-range identity (kk*11)>>5. K-steps padded to even, unrolled x2.
// ---------------------------------------------------------------------------
__global__ __launch_bounds__(256) void conv3x3_wmma(
    const float* __restrict__ in, const float* __restrict__ wgt,
    const float* __restrict__ bias, float* __restrict__ out,
    int Bn, int Cin, int Cout, int H, int W)
{
  const int lane = threadIdx.x & 31;
  const int wave = threadIdx.x >> 5;
  const int hi   = lane >> 4;
  const int col  = lane & 15;

  const int  K    = Cin * 9;
  const int  Ks   = (K + 3) >> 2;
  const int  Ks2  = (Ks + 1) & ~1;          // padded to even
  const int  HWp  = H * W;
  const int  Npix = Bn * HWp;
  const int  ntN  = Npix >> 4;
  const int  ntM  = (Cout + 15) >> 4;
  const long ntiles  = (long)ntM * ntN;
  const long wstride = (long)gridDim.x * (blockDim.x >> 5);

  for (long tile = (long)blockIdx.x * (blockDim.x >> 5) + wave;
       tile < ntiles; tile += wstride) {
    const int mt = (int)(tile / ntN);
    const int nt = (int)(tile % ntN);
    const int m0 = mt << 4;
    // per-lane output pixel (column of the tile)
    const int n = (nt << 4) + col;
    const int x = n % W;
    const int y = (n / W) % H;
    const int b = n / HWp;
    const int boff = b * Cin * HWp;            // 32-bit batch base offset
    // per-lane A row (clamped; invalid rows zeroed by select)
    const int  ma   = m0 + col;
    const bool mok  = (ma < Cout);
    const int  mrow = (mok ? ma : 0) * K;      // 32-bit weight row offset

    // branch-free B gather: clamped 32-bit offset, selected value
    auto gatherB = [&](int ci_, int kk_) -> float {
      const int t  = (kk_ * 11) >> 5;              // kk_/3 for kk_ in [0,8]
      const int iy = y + t - 1;
      const int ix = x + (kk_ - t * 3) - 1;
      const bool ok = ((unsigned)iy < (unsigned)H) &
                      ((unsigned)ix < (unsigned)W) & (ci_ < Cin);
      const int iyc = min(max(iy, 0), H - 1);
      const int ixc = min(max(ix, 0), W - 1);
      const int cic = min(ci_, Cin - 1);
      const float v = in[(unsigned)(boff + cic * HWp + iyc * W + ixc)];
      return ok ? v : 0.f;
    };
    auto gatherA = [&](int ci_, int kk_) -> float {
      const bool ok = mok & (ci_ < Cin);
      const int cic = min(ci_, Cin - 1);
      const float v = wgt[(unsigned)(mrow + cic * 9 + kk_)];
      return ok ? v : 0.f;
    };

    // per-lane (ci, kk) state for k0 = 4*ks + 2*hi
    int ci = 0, kk = hi << 1;                      // 2*hi < 9
    v8f acc = {};
    for (int ks = 0; ks < Ks2; ks += 2) {
      // ---- step 0 fragments
      int kk1 = kk + 1, ci1 = ci;
      if (kk1 >= 9) { kk1 -= 9; ++ci1; }
      const float a00 = gatherA(ci, kk);
      const float a01 = gatherA(ci1, kk1);
      const float b00 = gatherB(ci, kk);
      const float b01 = gatherB(ci1, kk1);
      kk += 4; if (kk >= 9) { kk -= 9; ++ci; }
      // ---- step 1 fragments
      int kk1b = kk + 1, ci1b = ci;
      if (kk1b >= 9) { kk1b -= 9; ++ci1b; }
      const float a10 = gatherA(ci, kk);
      const float a11 = gatherA(ci1b, kk1b);
      const float b10 = gatherB(ci, kk);
      const float b11 = gatherB(ci1b, kk1b);
      kk += 4; if (kk >= 9) { kk -= 9; ++ci; }

      v2f av0 = {a00, a01}, bv0 = {b00, b01};
      acc = __builtin_amdgcn_wmma_f32_16x16x4_f32(
          false, av0, false, bv0, (short)0, acc, false, false);
      v2f av1 = {a10, a11}, bv1 = {b10, b11};
      acc = __builtin_amdgcn_wmma_f32_16x16x4_f32(
          false, av1, false, bv1, (short)0, acc, false, false);
    }
    for (int r = 0; r < 8; ++r) {
      const int co = m0 + r + (hi << 3);
      if (co < Cout) {
        out[(unsigned)((b * Cout + co) * HWp + y * W + x)] = acc[r] + bias[co];
      }
    }
  }
}

// ---------------------------------------------------------------------------
// WMMA GEMM for the deformable conv: A[M x K] (weights), Bm[K x N] (sampled
// im2col, coalesced), out NCHW (B,M,4,4), fused ReLU. M,K,N are exact
// multiples of the tile sizes, so the loop is guard-free. 32-bit offsets.
// ---------------------------------------------------------------------------
__global__ __launch_bounds__(256) void gemm_wmma_dconv(
    const float* __restrict__ A, const float* __restrict__ Bm,
    float* __restrict__ out, int M, int K, int N)
{
  const int lane = threadIdx.x & 31;
  const int wave = threadIdx.x >> 5;
  const int hi   = lane >> 4;
  const int col  = lane & 15;

  const int  Ks  = K >> 2;
  const int  ntN = N >> 4;
  const int  ntM = M >> 4;
  const long ntiles  = (long)ntM * ntN;
  const long wstride = (long)gridDim.x * (blockDim.x >> 5);

  for (long tile = (long)blockIdx.x * (blockDim.x >> 5) + wave;
       tile < ntiles; tile += wstride) {
    const int mt = (int)(tile / ntN);
    const int nt = (int)(tile % ntN);
    const int m0 = mt << 4;
    const int n  = (nt << 4) + col;
    const int arow = (m0 + col) * K;           // 32-bit offsets

    v8f acc = {};
#pragma unroll 2
    for (int ks = 0; ks < Ks; ++ks) {
      const int k0 = (ks << 2) + (hi << 1);
      const float a0 = A[(unsigned)(arow + k0)];
      const float a1 = A[(unsigned)(arow + k0 + 1)];
      const float b0 = Bm[(unsigned)(k0 * N + n)];
      const float b1 = Bm[(unsigned)((k0 + 1) * N + n)];
      v2f av = {a0, a1};
      v2f bv = {b0, b1};
      acc = __builtin_amdgcn_wmma_f32_16x16x4_f32(
          false, av, false, bv, (short)0, acc, false, false);
    }
    const int bb  = n >> 4;
    const int pix = n & 15;
    for (int r = 0; r < 8; ++r) {
      const int m = m0 + r + (hi << 3);
      out[(unsigned)((bb * M + m) * 16 + pix)] = fmaxf(acc[r], 0.f);
    }
  }
}

// ---------------------------------------------------------------------------
// Per-channel batchnorm statistics, folded into scale/shift:
//   bn(x) = x*scale[c] + shift[c]
// ---------------------------------------------------------------------------
__global__ __launch_bounds__(256) void bn_stats(
    const float* __restrict__ x, const float* __restrict__ g,
    const float* __restrict__ bt, float* __restrict__ scale,
    float* __restrict__ shift, int C, int Bn, int HW)
{
  const int c   = blockIdx.x;
  const int per = Bn * HW;
  float s = 0.f, s2 = 0.f;
  for (int e = threadIdx.x; e < per; e += blockDim.x) {
    const int b = e / HW, pix = e - b * HW;
    const float v = x[(unsigned)((b * C + c) * HW + pix)];
    s += v;
    s2 += v * v;
  }
  __shared__ float ls[256];
  __shared__ float ls2[256];
  ls[threadIdx.x]  = s;
  ls2[threadIdx.x] = s2;
  __syncthreads();
  for (int st = 128; st > 0; st >>= 1) {
    if ((int)threadIdx.x < st) {
      ls[threadIdx.x]  += ls[threadIdx.x + st];
      ls2[threadIdx.x] += ls2[threadIdx.x + st];
    }
    __syncthreads();
  }
  if (threadIdx.x == 0) {
    const float inv  = 1.f / (float)per;
    const float mean = ls[0] * inv;
    const float var  = ls2[0] * inv - mean * mean;
    const float r    = rsqrtf(var + BN_EPS);
    const float sc   = g[c] * r;
    scale[c] = sc;
    shift[c] = bt[c] - mean * sc;
  }
}

// ---------------------------------------------------------------------------
// Fused affine-BN + 2x2 maxpool + ReLU.
// ---------------------------------------------------------------------------
__global__ void bn_pool_relu(const float* __restrict__ x,
                             const float* __restrict__ scale,
                             const float* __restrict__ shift,
                             float* __restrict__ y, int Bn, int C, int H, int W)
{
  const int Ho = H >> 1, Wo = W >> 1;
  const int total = Bn * C * Ho * Wo;
  const int i = blockIdx.x * blockDim.x + threadIdx.x;
  if (i >= total) return;
  const int xo = i % Wo;
  int t = i / Wo;
  const int yo = t % Ho; t /= Ho;
  const int c  = t % C;
  const int b  = t / C;
  const float sc = scale[c], sh = shift[c];
  const unsigned base = (unsigned)((b * C + c) * H * W + (2 * yo) * W + 2 * xo);
  const float v0 = x[base] * sc + sh;
  const float v1 = x[base + 1] * sc + sh;
  const float v2 = x[base + W] * sc + sh;
  const float v3 = x[base + W + 1] * sc + sh;
  const float m = fmaxf(fmaxf(v0, v1), fmaxf(v2, v3));
  y[i] = fmaxf(m, 0.f);
}

// ---------------------------------------------------------------------------
// Deformable-conv bilinear sampling. Produces G[k][ncol] with
// k = ci*9 + tap, ncol = b*16 + pix (so GEMM B loads are coalesced).
// Reproduces the reference: offset channel reorder (even->x, odd->y),
// p = p0 + p_n + off, border floor, clip to [0,5], bilinear on 6x6 padded.
// ---------------------------------------------------------------------------
__global__ void deform_sample(const float* __restrict__ x,   // (128,256,4,4)
                              const float* __restrict__ off, // (128,18,4,4)
                              float* __restrict__ G)         // [2304][2048]
{
  const int tid = blockIdx.x * blockDim.x + threadIdx.x;
  if (tid >= 2304 * 2048) return;
  const int pix = tid & 15;
  const int b   = (tid >> 4) & 127;
  const int k   = tid >> 11;
  const int t   = k % 9;
  const int ci  = k / 9;
  const int yq  = pix >> 2, xq = pix & 3;

  const unsigned obase = (unsigned)(b * 18 * 16);
  const float ox = off[obase + (2 * t) * 16 + pix];     // even ch -> x (rows)
  const float oy = off[obase + (2 * t + 1) * 16 + pix]; // odd ch  -> y (cols)

  float px = (float)(yq + 1) + (float)(t / 3 - 1) + ox;
  float py = (float)(xq + 1) + (float)(t % 3 - 1) + oy;
  if (px < 1.f || px > 4.f) px = floorf(px);
  if (py < 1.f || py > 4.f) py = floorf(py);
  px = fminf(fmaxf(px, 0.f), 5.f);
  py = fminf(fmaxf(py, 0.f), 5.f);

  const float fx = floorf(px), fy = floorf(py);
  const float qltx = fx;
  const float qlty = fy;
  const float qrbx = fminf(fx + 1.f, 5.f);
  const float qrby = fminf(fy + 1.f, 5.f);

  const float glt = (1.f + (qltx - px)) * (1.f + (qlty - py));
  const float grb = (1.f - (qrbx - px)) * (1.f - (qrby - py));
  const float glb = (1.f + (qltx - px)) * (1.f - (qrby - py));
  const float grt = (1.f - (qrbx - px)) * (1.f + (qlty - py));

  const unsigned xbase = (unsigned)((b * 256 + ci) * 16);
  auto S = [&](float qx, float qy) -> float {
    const int ix = (int)qx - 1, jy = (int)qy - 1;  // unpad
    const bool ok = (ix >= 0) & (ix < 4) & (jy >= 0) & (jy < 4);
    const int ixc = min(max(ix, 0), 3);
    const int jyc = min(max(jy, 0), 3);
    const float v = x[xbase + ixc * 4 + jyc];      // always-valid address
    return ok ? v : 0.f;
  };
  const float val = glt * S(qltx, qlty) + grb * S(qrbx, qrby) +
                    glb * S(qltx, qrby) + grt * S(qrbx, qlty);
  G[tid] = val;
}

// ---------------------------------------------------------------------------
// bn5 apply + spatial mean: feat[b,c] = mean_pix(bn(dconv)).
// ---------------------------------------------------------------------------
__global__ void bn_mean_feat(const float* __restrict__ dconv,
                             const float* __restrict__ scale,
                             const float* __restrict__ shift,
                             float* __restrict__ feat)
{
  const int i = blockIdx.x * blockDim.x + threadIdx.x;  // b*256 + c
  if (i >= 128 * 256) return;
  const int c = i & 255;
  const float* __restrict__ p = dconv + (unsigned)i * 16;
  float s = 0.f;
  for (int t = 0; t < 16; ++t) s += p[t];
  feat[i] = (s * scale[c]) * (1.f / 16.f) + shift[c];
}

// ---------------------------------------------------------------------------
// FC (256 -> 250) + log_softmax, one block per batch row.
// ---------------------------------------------------------------------------
__global__ __launch_bounds__(256) void fc_logsoftmax(
    const float* __restrict__ feat, const float* __restrict__ fw,
    const float* __restrict__ fb, float* __restrict__ out)
{
  const int b = blockIdx.x;
  const int t = threadIdx.x;
  __shared__ float f[256];
  __shared__ float red[256];
  f[t] = feat[b * 256 + t];
  __syncthreads();

  float logit = -1e30f;
  if (t < 250) {
    float s = fb[t];
    const float* __restrict__ wr = fw + (unsigned)(t * 256);
    for (int c = 0; c < 256; ++c) s += f[c] * wr[c];
    logit = s;
  }
  red[t] = logit;
  __syncthreads();
  for (int st = 128; st > 0; st >>= 1) {
    if (t < st) red[t] = fmaxf(red[t], red[t + st]);
    __syncthreads();
  }
  const float mx = red[0];
  __syncthreads();
  red[t] = (t < 250) ? expf(logit - mx) : 0.f;
  __syncthreads();
  for (int st = 128; st > 0; st >>= 1) {
    if (t < st) red[t] += red[t + st];
    __syncthreads();
  }
  const float lse = logf(red[0]);
  if (t < 250) out[b * 250 + t] = logit - mx - lse;
}

// ---------------------------------------------------------------------------
extern "C" void kernel_launch(void* const* d_in, const int* in_sizes, int n_in,
                              void* d_out, int out_size, void* d_ws,
                              size_t ws_size, hipStream_t stream)
{
  const float* s    = (const float*)d_in[0];
  const float* c1w  = (const float*)d_in[1];
  const float* c1b  = (const float*)d_in[2];
  const float* g1   = (const float*)d_in[3];
  const float* bb1  = (const float*)d_in[4];
  const float* c2w  = (const float*)d_in[5];
  const float* c2b  = (const float*)d_in[6];
  const float* g2   = (const float*)d_in[7];
  const float* bb2  = (const float*)d_in[8];
  const float* c3w  = (const float*)d_in[9];
  const float* c3b  = (const float*)d_in[10];
  const float* g3   = (const float*)d_in[11];
  const float* bb3  = (const float*)d_in[12];
  const float* c4w  = (const float*)d_in[13];
  const float* c4b  = (const float*)d_in[14];
  const float* g4   = (const float*)d_in[15];
  const float* bb4  = (const float*)d_in[16];
  const float* offw = (const float*)d_in[17];
  const float* offb = (const float*)d_in[18];
  const float* dw   = (const float*)d_in[19];
  const float* g5   = (const float*)d_in[20];
  const float* bb5  = (const float*)d_in[21];
  const float* fw   = (const float*)d_in[22];
  const float* fb   = (const float*)d_in[23];
  float* out = (float*)d_out;

  // Workspace layout (floats). conv region is reused for G/dconv/feat after
  // the last pooling consumes conv4's raw output.
  float* ws    = (float*)d_ws;
  float* conv  = ws;                    // 33,554,432 (conv1 raw out, max)
  float* poolA = conv + 33554432;       //  8,388,608
  float* poolB = poolA + 8388608;       //  8,388,608
  float* scale = poolB + 8388608;       //        256
  float* shift = scale + 256;           //        256
  float* offo  = shift + 256;           //     36,864 (B,18,4,4)
  float* G     = conv;                  //  4,718,592 (overlaps dead conv buf)
  float* dcv   = conv + 4718592;        //    524,288
  float* feat  = dcv + 524288;          //     32,768

  auto cblocks = [](long cout_t, long npix) {
    const long tiles = ((cout_t + 15) / 16) * (npix / 16);
    return (int)((tiles + 7) / 8);  // 8 waves per 256-thread block
  };

  // ---- stage 1: conv1 (1->64 @64x64), bn1, pool, relu -> poolA (64@32x32)
  conv3x3_wmma<<<cblocks(64, 128L * 64 * 64), 256, 0, stream>>>(
      s, c1w, c1b, conv, 128, 1, 64, 64, 64);
  bn_stats<<<64, 256, 0, stream>>>(conv, g1, bb1, scale, shift, 64, 128, 64 * 64);
  bn_pool_relu<<<(int)((128L * 64 * 32 * 32 + 255) / 256), 256, 0, stream>>>(
      conv, scale, shift, poolA, 128, 64, 64, 64);

  // ---- stage 2: conv2 (64->128 @32x32), bn2, pool, relu -> poolB (128@16x16)
  conv3x3_wmma<<<cblocks(128, 128L * 32 * 32), 256, 0, stream>>>(
      poolA, c2w, c2b, conv, 128, 64, 128, 32, 32);
  bn_stats<<<128, 256, 0, stream>>>(conv, g2, bb2, scale, shift, 128, 128, 32 * 32);
  bn_pool_relu<<<(int)((128L * 128 * 16 * 16 + 255) / 256), 256, 0, stream>>>(
      conv, scale, shift, poolB, 128, 128, 32, 32);

  // ---- stage 3: conv3 (128->256 @16x16), bn3, pool, relu -> poolA (256@8x8)
  conv3x3_wmma<<<cblocks(256, 128L * 16 * 16), 256, 0, stream>>>(
      poolB, c3w, c3b, conv, 128, 128, 256, 16, 16);
  bn_stats<<<256, 256, 0, stream>>>(conv, g3, bb3, scale, shift, 256, 128, 16 * 16);
  bn_pool_relu<<<(int)((128L * 256 * 8 * 8 + 255) / 256), 256, 0, stream>>>(
      conv, scale, shift, poolA, 128, 256, 16, 16);

  // ---- stage 4: conv4 (256->256 @8x8), bn4, pool, relu -> poolB (256@4x4)
  conv3x3_wmma<<<cblocks(256, 128L * 8 * 8), 256, 0, stream>>>(
      poolA, c4w, c4b, conv, 128, 256, 256, 8, 8);
  bn_stats<<<256, 256, 0, stream>>>(conv, g4, bb4, scale, shift, 256, 128, 8 * 8);
  bn_pool_relu<<<(int)((128L * 256 * 4 * 4 + 255) / 256), 256, 0, stream>>>(
      conv, scale, shift, poolB, 128, 256, 8, 8);

  // ---- offset conv (256->18 @4x4), with bias, no BN/ReLU
  conv3x3_wmma<<<cblocks(18, 128L * 4 * 4), 256, 0, stream>>>(
      poolB, offw, offb, offo, 128, 256, 18, 4, 4);

  // ---- deformable sampling -> G [2304][2048]
  deform_sample<<<(2304 * 2048) / 256, 256, 0, stream>>>(poolB, offo, G);

  // ---- deformable conv GEMM (256 x 2304 x 2048) + fused ReLU -> dcv
  gemm_wmma_dconv<<<cblocks(256, 2048), 256, 0, stream>>>(
      dw, G, dcv, 256, 2304, 2048);

  // ---- bn5 stats, then BN + spatial mean -> feat (128,256)
  bn_stats<<<256, 256, 0, stream>>>(dcv, g5, bb5, scale, shift, 256, 128, 16);
  bn_mean_feat<<<(128 * 256) / 256, 256, 0, stream>>>(dcv, scale, shift, feat);

  // ---- FC + log_softmax -> d_out (128,250)
  fc_logsoftmax<<<128, 256, 0, stream>>>(feat, fw, fb, out);
}